// GraphModel_51376398795539
// MI455X (gfx1250) — compile-verified
//
#include <hip/hip_runtime.h>

#define N_NODES 100000
#define NUM_E   1600000
#define HDIM    128
#define LAYERS  4
#define OUTD    64
#define NROOTS  1024
#define WT_STRIDE 130   // padded LDS stride for transposed W: even (b64 align) + conflict-free-ish

typedef float v2f __attribute__((ext_vector_type(2)));
typedef float v8f __attribute__((ext_vector_type(8)));

// ---------------- degree / normalization ----------------

__global__ void init_deg_kernel(float* __restrict__ deg, int n) {
    int i = blockIdx.x * blockDim.x + threadIdx.x;
    if (i < n) deg[i] = 1.0f;   // self-loop contributes 1
}

__global__ void accum_deg_kernel(const int* __restrict__ dst, float* __restrict__ deg, int e) {
    int i = blockIdx.x * blockDim.x + threadIdx.x;
    if (i < e) atomicAdd(&deg[dst[i]], 1.0f);
}

__global__ void dinv_kernel(float* __restrict__ deg, int n) {
    int i = blockIdx.x * blockDim.x + threadIdx.x;
    if (i < n) deg[i] = rsqrtf(fmaxf(deg[i], 1.0f));
}

// ---------------- fp32 WMMA GEMM: C[M x 128] = A[M x 128] * W[128 x 128] + bias ----------------
// block = 256 threads (8 waves). Each block covers 128 rows; each wave a 16x128 strip.
// W staged TRANSPOSED in LDS (Wt[col*130 + k]) so each B fragment is one ds_load_b64.
// Uses V_WMMA_F32_16X16X4_F32, accumulating over 32 K-steps.

__global__ void gemm_h128_kernel(const float* __restrict__ A,
                                 const float* __restrict__ W,
                                 const float* __restrict__ bias,
                                 float* __restrict__ C, int M) {
    extern __shared__ float Wt[];                 // 130*128 floats = 66560 B
    const int tid = threadIdx.x;

    {   // cooperative transpose-stage of W into LDS: Wt[c*WT_STRIDE + k] = W[k*128 + c]
        const float4* Wg = (const float4*)W;
        for (int i = tid; i < (HDIM * HDIM) / 4; i += 256) {
            int k  = i >> 5;             // 32 float4 per row of W
            int c4 = (i & 31) * 4;
            float4 v = Wg[i];
            Wt[(c4 + 0) * WT_STRIDE + k] = v.x;
            Wt[(c4 + 1) * WT_STRIDE + k] = v.y;
            Wt[(c4 + 2) * WT_STRIDE + k] = v.z;
            Wt[(c4 + 3) * WT_STRIDE + k] = v.w;
        }
    }
    __syncthreads();

    const int wave = tid >> 5;
    const int lane = tid & 31;
    const int mrow = lane & 15;                   // row/col within 16-tile
    const int kset = lane >> 4;                   // 0 -> K+0/K+1, 1 -> K+2/K+3
    const int rowBase = blockIdx.x * 128 + wave * 16;

    int arow = rowBase + mrow;
    if (arow >= M) arow = M - 1;                  // clamp: garbage rows never stored
    const float* __restrict__ Arow = A + (size_t)arow * HDIM;
    __builtin_prefetch(Arow, 0, 0);               // global_prefetch_b8

    v8f acc[8];
    const v8f zero = {};
#pragma unroll
    for (int t = 0; t < 8; ++t) acc[t] = zero;

    for (int k = 0; k < HDIM; k += 4) {
        v2f a;
        a.x = Arow[k + 2 * kset + 0];
        a.y = Arow[k + 2 * kset + 1];
#pragma unroll
        for (int t = 0; t < 8; ++t) {
            // B frag: cols t*16+mrow, K pair (k+2*kset, k+1+2*kset) -> contiguous in Wt
            v2f b = *(const v2f*)(Wt + (t * 16 + mrow) * WT_STRIDE + k + 2 * kset);
            acc[t] = __builtin_amdgcn_wmma_f32_16x16x4_f32(
                false, a, false, b, (short)0, acc[t], false, false);
        }
    }

#pragma unroll
    for (int t = 0; t < 8; ++t) {
        float bv = bias[t * 16 + mrow];
#pragma unroll
        for (int r = 0; r < 8; ++r) {
            int row = rowBase + 8 * kset + r;     // C/D layout: lanes>=16 hold rows 8..15
            if (row < M) C[(size_t)row * HDIM + t * 16 + mrow] = acc[t][r] + bv;
        }
    }
}

// ---------------- self-loop init: agg = lin * dinv^2 ----------------

__global__ void init_agg_kernel(const float* __restrict__ lin, const float* __restrict__ dinv,
                                float* __restrict__ agg, int n) {
    int tid = blockIdx.x * blockDim.x + threadIdx.x;   // over n*32 float4 chunks
    if (tid >= n * 32) return;
    int node = tid >> 5;
    float w = dinv[node]; w *= w;
    float4 v = ((const float4*)lin)[tid];
    float4 o = {v.x * w, v.y * w, v.z * w, v.w * w};
    ((float4*)agg)[tid] = o;
}

// ---------------- edge scatter: agg[dst] += lin[src] * dinv[src]*dinv[dst] ----------------

__global__ void scatter_kernel(const float* __restrict__ lin,
                               const int* __restrict__ src, const int* __restrict__ dst,
                               const float* __restrict__ dinv, float* __restrict__ agg, int e) {
    int tid = blockIdx.x * blockDim.x + threadIdx.x;   // 32 lanes per edge (float4 each)
    int ed = tid >> 5;
    int c  = tid & 31;
    if (ed >= e) return;
    int s = src[ed], d = dst[ed];
    float w = dinv[s] * dinv[d];
    float4 v = ((const float4*)(lin + (size_t)s * HDIM))[c];
    float* out = agg + (size_t)d * HDIM + c * 4;
    atomicAdd(out + 0, v.x * w);
    atomicAdd(out + 1, v.y * w);
    atomicAdd(out + 2, v.z * w);
    atomicAdd(out + 3, v.w * w);
}

// ---------------- relu + residual + layernorm (one wave per node) ----------------

__global__ void post_ln_kernel(float* __restrict__ h, const float* __restrict__ agg,
                               const float* __restrict__ gamma, const float* __restrict__ beta,
                               int n) {
    int wave = threadIdx.x >> 5, lane = threadIdx.x & 31;
    int node = blockIdx.x * (blockDim.x >> 5) + wave;
    if (node >= n) return;

    const float4* a4 = (const float4*)(agg + (size_t)node * HDIM);
    float4* h4 = (float4*)(h + (size_t)node * HDIM);
    float4 a = a4[lane];
    float4 hv = h4[lane];
    float4 v;
    v.x = hv.x + fmaxf(a.x, 0.0f);
    v.y = hv.y + fmaxf(a.y, 0.0f);
    v.z = hv.z + fmaxf(a.z, 0.0f);
    v.w = hv.w + fmaxf(a.w, 0.0f);

    float s1 = v.x + v.y + v.z + v.w;
    float s2 = v.x * v.x + v.y * v.y + v.z * v.z + v.w * v.w;
#pragma unroll
    for (int m = 16; m > 0; m >>= 1) {            // wave32 butterfly reduction
        s1 += __shfl_xor(s1, m, 32);
        s2 += __shfl_xor(s2, m, 32);
    }
    float mu   = s1 * (1.0f / HDIM);
    float var  = s2 * (1.0f / HDIM) - mu * mu;
    float rstd = rsqrtf(var + 1e-5f);

    float4 g = ((const float4*)gamma)[lane];
    float4 b = ((const float4*)beta)[lane];
    float4 o;
    o.x = (v.x - mu) * rstd * g.x + b.x;
    o.y = (v.y - mu) * rstd * g.y + b.y;
    o.z = (v.z - mu) * rstd * g.z + b.z;
    o.w = (v.w - mu) * rstd * g.w + b.w;
    h4[lane] = o;
}

// ---------------- readout: out[r] = h[root_idx[r]] @ W_out + b_out (WMMA) ----------------
// 1024 roots, 64 cols. Block = 8 waves; each wave 16 rows x 64 cols (4 tiles). Grid = 8.
// W_out staged into LDS with GLOBAL_LOAD_ASYNC_TO_LDS_B128 (ASYNCcnt-tracked DMA path).

__global__ void readout_kernel(const float* __restrict__ h, const int* __restrict__ roots,
                               const float* __restrict__ Wout, const float* __restrict__ bout,
                               float* __restrict__ out) {
    extern __shared__ float Wl[];                 // 128*64 floats = 32 KB, LDS offset 0
    const int tid = threadIdx.x;
    {
        // async global->LDS copy: 16B per lane per issue; LDS dest offset is wave-relative,
        // and the dynamic-shared block is the only LDS allocation (offset 0).
        for (int i = tid; i < (HDIM * OUTD) / 4; i += 256) {
            unsigned int  lds_off = (unsigned int)(i * 16);
            unsigned long long gsrc = (unsigned long long)((const char*)Wout + (size_t)i * 16);
            asm volatile("global_load_async_to_lds_b128 %0, %1, off"
                         :: "v"(lds_off), "v"(gsrc)
                         : "memory");
        }
        asm volatile("s_wait_asynccnt 0x0" ::: "memory");
    }
    __syncthreads();

    const int wave = tid >> 5;
    const int lane = tid & 31;
    const int mrow = lane & 15;
    const int kset = lane >> 4;
    const int rowBase = (blockIdx.x * 8 + wave) * 16;   // exact: 8*8*16 = 1024

    int ridx = roots[rowBase + mrow];
    const float* __restrict__ Arow = h + (size_t)ridx * HDIM;

    v8f acc[4];
    const v8f zero = {};
#pragma unroll
    for (int t = 0; t < 4; ++t) acc[t] = zero;

    for (int k = 0; k < HDIM; k += 4) {
        v2f a;
        a.x = Arow[k + 2 * kset + 0];
        a.y = Arow[k + 2 * kset + 1];
#pragma unroll
        for (int t = 0; t < 4; ++t) {
            v2f b;
            b.x = Wl[(k + 2 * kset + 0) * OUTD + t * 16 + mrow];
            b.y = Wl[(k + 2 * kset + 1) * OUTD + t * 16 + mrow];
            acc[t] = __builtin_amdgcn_wmma_f32_16x16x4_f32(
                false, a, false, b, (short)0, acc[t], false, false);
        }
    }

#pragma unroll
    for (int t = 0; t < 4; ++t) {
        float bv = bout[t * 16 + mrow];
#pragma unroll
        for (int r = 0; r < 8; ++r) {
            int row = rowBase + 8 * kset + r;
            out[(size_t)row * OUTD + t * 16 + mrow] = acc[t][r] + bv;
        }
    }
}

// ---------------- host orchestration ----------------

extern "C" void kernel_launch(void* const* d_in, const int* in_sizes, int n_in,
                              void* d_out, int out_size, void* d_ws, size_t ws_size,
                              hipStream_t stream) {
    const float* x        = (const float*)d_in[0];
    const int*   edge     = (const int*)d_in[1];      // [2, E]
    const int*   roots    = (const int*)d_in[2];
    const float* W_embed  = (const float*)d_in[3];
    const float* b_embed  = (const float*)d_in[4];
    const float* W_layers = (const float*)d_in[5];    // [L,H,H]
    const float* b_layers = (const float*)d_in[6];    // [L,H]
    const float* ln_gamma = (const float*)d_in[7];
    const float* ln_beta  = (const float*)d_in[8];
    const float* W_out    = (const float*)d_in[9];
    const float* b_out    = (const float*)d_in[10];
    float* out = (float*)d_out;

    const int* src = edge;
    const int* dst = edge + NUM_E;

    // workspace carve
    float* dinv = (float*)d_ws;                                   // N
    float* h    = dinv + N_NODES;                                 // N*H
    float* lin  = h + (size_t)N_NODES * HDIM;                     // N*H
    float* agg  = lin + (size_t)N_NODES * HDIM;                   // N*H

    const int B = 256;
    const size_t gemmLds = (size_t)WT_STRIDE * HDIM * sizeof(float);

    // degree normalization
    init_deg_kernel<<<(N_NODES + B - 1) / B, B, 0, stream>>>(dinv, N_NODES);
    accum_deg_kernel<<<(NUM_E + B - 1) / B, B, 0, stream>>>(dst, dinv, NUM_E);
    dinv_kernel<<<(N_NODES + B - 1) / B, B, 0, stream>>>(dinv, N_NODES);

    // embed: h = x @ W_embed + b_embed
    const int gemmGrid = (N_NODES + 127) / 128;
    gemm_h128_kernel<<<gemmGrid, B, gemmLds, stream>>>(x, W_embed, b_embed, h, N_NODES);

    // GNN layers
    for (int l = 0; l < LAYERS; ++l) {
        const float* Wl_p = W_layers + (size_t)l * HDIM * HDIM;
        const float* bl_p = b_layers + (size_t)l * HDIM;
        const float* g_p  = ln_gamma + (size_t)l * HDIM;
        const float* be_p = ln_beta  + (size_t)l * HDIM;

        gemm_h128_kernel<<<gemmGrid, B, gemmLds, stream>>>(h, Wl_p, bl_p, lin, N_NODES);

        init_agg_kernel<<<((N_NODES * 32) + B - 1) / B, B, 0, stream>>>(lin, dinv, agg, N_NODES);

        long long sthreads = (long long)NUM_E * 32;
        scatter_kernel<<<(int)((sthreads + B - 1) / B), B, 0, stream>>>(
            lin, src, dst, dinv, agg, NUM_E);

        post_ln_kernel<<<(N_NODES + 7) / 8, B, 0, stream>>>(h, agg, g_p, be_p, N_NODES);
    }

    // readout on gathered roots (async-LDS staged weights)
    readout_kernel<<<NROOTS / 128, B, HDIM * OUTD * sizeof(float), stream>>>(
        h, roots, W_out, b_out, out);
}